// LorentzAggregator_10574209483386
// MI455X (gfx1250) — compile-verified
//
#include <hip/hip_runtime.h>
#include <hip/hip_bf16.h>
#include <math.h>

typedef __attribute__((ext_vector_type(2))) float v2f;
typedef __attribute__((ext_vector_type(8))) float v8f;

#define DD 64  // d+1 (attention dim), reference D1

// ---- order-preserving float <-> uint encoding for integer atomicMax ----
__device__ __forceinline__ unsigned fenc(float f) {
  int i = __float_as_int(f);
  return (i >= 0) ? ((unsigned)i ^ 0x80000000u) : ~(unsigned)i;
}
__device__ __forceinline__ float fdec(unsigned k) {
  int i = (k & 0x80000000u) ? (int)(k ^ 0x80000000u) : (int)~k;
  return __int_as_float(i);
}
__device__ __forceinline__ float bcast_f32(float v, int srcLane) {
  return __int_as_float(__builtin_amdgcn_readlane(__float_as_int(v), srcLane));
}

// ---------------- init: seed max-keys, zero z/agg/tanagg --------------------
__global__ __launch_bounds__(256) void k_init(unsigned* __restrict__ mkeys,
                                              float* __restrict__ zr,
                                              int N, long long zcount) {
  long long t = (long long)blockIdx.x * blockDim.x + threadIdx.x;
  if (t < N) mkeys[t] = 0x00800000u;  // fenc(-FLT_MAX)
  if (t < zcount) zr[t] = 0.0f;
}

// ---------------- per-node dot via V_WMMA_F32_16X16X4_F32 -------------------
// One wave = 16 nodes. A: 16 nodes x 4 spatial dims per step (dims 1..63, the
// pad slot's A value is clamped in-bounds and killed by a zero B weight, so all
// loads are unconditional and EXEC never changes). B: weight slice replicated
// across all 16 columns -> every D column holds the full 63-dim dot.
// Two independent accumulator chains (even/odd k) break the serial D->C chain.
__global__ __launch_bounds__(256) void k_node_dot_wmma(
    const float* __restrict__ x, const float* __restrict__ w,
    float* __restrict__ araw, int N) {
  const int lane = threadIdx.x & 31;
  const int wave = (blockIdx.x * blockDim.x + threadIdx.x) >> 5;
  const int base = wave * 16;
  if (base >= N) return;                 // wave-uniform: EXEC stays all-1s
  const int half = lane >> 4;            // 0: (K0,K1), 1: (K2,K3)
  const int mrow = lane & 15;
  const int node = min(base + mrow, N - 1);   // branch-free ragged-tail clamp
  const float* __restrict__ xr = x + (size_t)node * DD;

  v8f c0 = {}, c1 = {};
#pragma unroll
  for (int k = 0; k < 16; ++k) {
    const int d0 = 1 + 4 * k + 2 * half;      // spatial dims d0, d0+1 (d0<=63)
    const int wd = 4 * k + 2 * half;          // attn_w index = dim-1 (wd<=62)
    v2f aop, bop;
    aop.x = xr[d0];
    aop.y = xr[(d0 + 1 <= DD - 1) ? (d0 + 1) : (DD - 1)];   // clamped pad slot
    bop.x = w[wd];
    const float wy = w[(wd + 1 < DD - 1) ? (wd + 1) : (DD - 2)];
    bop.y = (wd + 1 < DD - 1) ? wy : 0.0f;    // zero weight kills pad product
    if (k & 1) {
      c1 = __builtin_amdgcn_wmma_f32_16x16x4_f32(false, aop, false, bop,
                                                 (short)0, c1, false, false);
    } else {
      c0 = __builtin_amdgcn_wmma_f32_16x16x4_f32(false, aop, false, bop,
                                                 (short)0, c0, false, false);
    }
  }
  // Columns identical; lane 0 holds rows 0..7, lane 16 holds rows 8..15.
  if (mrow == 0) {
#pragma unroll
    for (int j = 0; j < 8; ++j) {
      int v = base + 8 * half + j;
      if (v < N) araw[v] = c0[j] + c1[j];
    }
  }
}

// ------- per-node scale: tscale = arccosh(x0)/||s||, a *= tscale ------------
__global__ __launch_bounds__(256) void k_node_scale(const float* __restrict__ x,
                                                    float* __restrict__ a,
                                                    float* __restrict__ tscale,
                                                    int N) {
  int v = blockIdx.x * blockDim.x + threadIdx.x;
  if (v >= N) return;
  const float* __restrict__ xr = x + (size_t)v * DD;
  float x0 = xr[0];
  float ssq = 0.0f;
#pragma unroll
  for (int d = 1; d < DD; ++d) { float s = xr[d]; ssq = fmaf(s, s, ssq); }
  ssq = fmaxf(ssq, 1e-12f);
  float scale = acoshf(fmaxf(x0, 1.0f + 1e-7f)) * rsqrtf(ssq);
  tscale[v] = scale;
  a[v] *= scale;
}

// ---------------- edge pass 1: segment max of scores ------------------------
__global__ __launch_bounds__(256) void k_edge_max(const int* __restrict__ row,
                                                  const int* __restrict__ col,
                                                  const float* __restrict__ a,
                                                  unsigned* __restrict__ mkeys,
                                                  int E) {
  int e = blockIdx.x * blockDim.x + threadIdx.x;
  if (e >= E) return;
  int r = row[e], c = col[e];
  atomicMax(&mkeys[c], fenc(a[r] + a[c]));
}

// ---------------- edge pass 2: exp-sum (softmax denominator) ----------------
__global__ __launch_bounds__(256) void k_edge_expsum(const int* __restrict__ row,
                                                     const int* __restrict__ col,
                                                     const float* __restrict__ a,
                                                     const unsigned* __restrict__ mkeys,
                                                     float* __restrict__ z, int E) {
  int e = blockIdx.x * blockDim.x + threadIdx.x;
  if (e >= E) return;
  int r = row[e], c = col[e];
  float ex = __expf(a[r] + a[c] - fdec(mkeys[c]));
  atomicAdd(&z[c], ex);
}

// -------- edge pass 3: wave-cooperative weighted scatter (agg + tan) --------
// Each wave stages 32 edges (lane-per-edge), then iterates them with scalar
// (readlane) broadcast of {r, c, w, wt}. All 32 lanes cover dims {lane,
// lane+32}: the x[r] gather is a fully coalesced 128B load pair and each
// atomic's lanes hit 32 consecutive dwords of agg[c]/tanagg[c] (L2-resident).
__global__ __launch_bounds__(256) void k_edge_agg(
    const int* __restrict__ row, const int* __restrict__ col,
    const float* __restrict__ a, const unsigned* __restrict__ mkeys,
    const float* __restrict__ z, const float* __restrict__ tscale,
    const float* __restrict__ x, float* __restrict__ agg,
    float* __restrict__ tanagg, int E) {
  const int lane = threadIdx.x & 31;
  const int wave = (blockIdx.x * blockDim.x + threadIdx.x) >> 5;
  const int base = wave * 32;
  if (base >= E) return;

  const int e = base + lane;
  int r_e = 0, c_e = 0;
  float w_e = 0.0f, wt_e = 0.0f;
  if (e < E) {
    r_e = row[e];
    c_e = col[e];
    float wv = __expf(a[r_e] + a[c_e] - fdec(mkeys[c_e])) / fmaxf(z[c_e], 1e-16f);
    w_e = wv;
    wt_e = wv * tscale[r_e];
  }

  const int nval = min(32, E - base);
  for (int j = 0; j < nval; ++j) {
    const int   r  = __builtin_amdgcn_readlane(r_e, j);
    const int   c  = __builtin_amdgcn_readlane(c_e, j);
    const float wv = bcast_f32(w_e, j);
    const float wt = bcast_f32(wt_e, j);
    const float* __restrict__ xr = x + (size_t)r * DD;
    const float xlo = xr[lane];
    const float xhi = xr[lane + 32];
    float* __restrict__ ag = agg + (size_t)c * DD;
    float* __restrict__ tg = tanagg + (size_t)c * DD;
    atomicAdd(&ag[lane], wv * xlo);
    atomicAdd(&ag[lane + 32], wv * xhi);
    if (lane != 0) atomicAdd(&tg[lane], wt * xlo);   // tangent time-coord is 0
    atomicAdd(&tg[lane + 32], wt * xhi);
  }
}

// ---------------- finalize: fallback, normalize, sheet correction -----------
__global__ __launch_bounds__(256) void k_finalize(const float* __restrict__ agg,
                                                  const float* __restrict__ tanagg,
                                                  float* __restrict__ out, int N) {
  int v = blockIdx.x * blockDim.x + threadIdx.x;
  if (v >= N) return;
  const float* __restrict__ ag = agg + (size_t)v * DD;
  const float* __restrict__ tg = tanagg + (size_t)v * DD;

  float A[DD];
  float sp = 0.0f;
#pragma unroll
  for (int d = 0; d < DD; ++d) {
    A[d] = ag[d];
    if (d > 0) sp = fmaf(A[d], A[d], sp);
  }
  float nsq = A[0] * A[0] - sp;                       // -mink(agg,agg)

  if (nsq <= 1e-8f) {                                 // tangent-space fallback
    float vnsq = 0.0f;
#pragma unroll
    for (int d = 1; d < DD; ++d) { float t = tg[d]; vnsq = fmaf(t, t, vnsq); }
    float vn = sqrtf(fmaxf(vnsq, 1e-15f));
    float sh = sinhf(vn) / vn;
    A[0] = coshf(vn);
#pragma unroll
    for (int d = 1; d < DD; ++d) A[d] = sh * tg[d];
  }

  float sp2 = 0.0f;
#pragma unroll
  for (int d = 1; d < DD; ++d) sp2 = fmaf(A[d], A[d], sp2);
  float nsq2 = A[0] * A[0] - sp2;
  float inv = 1.0f / fmaxf(sqrtf(fmaxf(nsq2, 0.0f)), 1e-12f);
  float sgn = (A[0] * inv <= 0.0f) ? -1.0f : 1.0f;    // upper-sheet correction
  float s = inv * sgn;
  float* __restrict__ o = out + (size_t)v * DD;
#pragma unroll
  for (int d = 0; d < DD; ++d) o[d] = A[d] * s;
}

extern "C" void kernel_launch(void* const* d_in, const int* in_sizes, int n_in,
                              void* d_out, int out_size, void* d_ws, size_t ws_size,
                              hipStream_t stream) {
  const float* x    = (const float*)d_in[0];
  const int*   ei   = (const int*)d_in[1];
  const float* attn = (const float*)d_in[2];

  const int D = in_sizes[2] + 1;        // = 64
  const int N = in_sizes[0] / D;
  const int E = in_sizes[1] / 2;
  const int* row = ei;
  const int* col = ei + E;

  // workspace: a[N] | tscale[N] | mkeys[N] | z[N] | agg[N*64] | tanagg[N*64]
  float*    a      = (float*)d_ws;
  float*    tscale = a + N;
  unsigned* mkeys  = (unsigned*)(tscale + N);
  float*    z      = (float*)(mkeys + N);
  float*    agg    = z + N;
  float*    tanagg = agg + (size_t)N * DD;
  const long long zcount = (long long)N * (2 * DD + 1);  // z + agg + tanagg

  const dim3 blk(256);
  k_init<<<(unsigned)((zcount + 255) / 256), blk, 0, stream>>>(mkeys, z, N, zcount);

  int nodeWaves = (N + 15) / 16;        // 16 nodes per wave32
  k_node_dot_wmma<<<(nodeWaves + 7) / 8, blk, 0, stream>>>(x, attn, a, N);
  k_node_scale<<<(N + 255) / 256, blk, 0, stream>>>(x, a, tscale, N);

  const int gE = (E + 255) / 256;
  k_edge_max   <<<gE, blk, 0, stream>>>(row, col, a, mkeys, E);
  k_edge_expsum<<<gE, blk, 0, stream>>>(row, col, a, mkeys, z, E);

  int edgeWaves = (E + 31) / 32;        // 32 edges per wave
  k_edge_agg<<<(edgeWaves + 7) / 8, blk, 0, stream>>>(row, col, a, mkeys, z,
                                                      tscale, x, agg, tanagg, E);

  k_finalize<<<(N + 255) / 256, blk, 0, stream>>>(agg, tanagg, (float*)d_out, N);
}